// plpLayer_79594333929560
// MI455X (gfx1250) — compile-verified
//
#include <hip/hip_runtime.h>

constexpr int N = 2048;
constexpr int F = 256;
constexpr int H = 32;
constexpr int C = 8;
constexpr float ALPHA = 0.9f;
constexpr float BETA  = 0.1f;
constexpr float ONE_M_EPS = 1.0f - 1e-6f;

typedef __attribute__((ext_vector_type(2))) float v2f;
typedef __attribute__((ext_vector_type(8))) float v8f;

// ---------------------------------------------------------------------------
// Kernel 1: pLabel = softmax(relu(x@W1 + b1) @ W2 + b2)
// One wave (32 lanes) computes a 16-row tile of h via V_WMMA_F32_16X16X4_F32.
// 4 waves / block -> 32 blocks cover N=2048 rows.
// ---------------------------------------------------------------------------
__global__ __launch_bounds__(128) void mlp_softmax_kernel(
    const float* __restrict__ x, const float* __restrict__ W1,
    const float* __restrict__ b1, const float* __restrict__ W2,
    const float* __restrict__ b2, float* __restrict__ P0)
{
    __shared__ float hlds[4][16][33];  // padded to dodge bank conflicts

    const int wave = threadIdx.x >> 5;
    const int lane = threadIdx.x & 31;
    const int row0 = (blockIdx.x * 4 + wave) * 16;

    // A-fragment addressing (ISA 7.12.2, 32-bit A 16x4):
    //   lanes 0-15: M=lane, regs = {K+0, K+1}; lanes 16-31: M=lane-16, regs = {K+2, K+3}
    const int mrow = lane & 15;
    const int kh   = (lane >> 4) << 1;   // 0 or 2
    const int ncol = lane & 15;          // B/D column

    const float* xr = x + (size_t)(row0 + mrow) * F;

    v8f acc0 = {};  // output cols 0..15
    v8f acc1 = {};  // output cols 16..31
    for (int kb = 0; kb < F; kb += 4) {
        v2f a;
        a[0] = xr[kb + kh];
        a[1] = xr[kb + kh + 1];
        v2f bb0, bb1;
        bb0[0] = W1[(kb + kh)     * H + ncol];
        bb0[1] = W1[(kb + kh + 1) * H + ncol];
        bb1[0] = W1[(kb + kh)     * H + ncol + 16];
        bb1[1] = W1[(kb + kh + 1) * H + ncol + 16];
        acc0 = __builtin_amdgcn_wmma_f32_16x16x4_f32(false, a, false, bb0,
                                                     (short)0, acc0, false, false);
        acc1 = __builtin_amdgcn_wmma_f32_16x16x4_f32(false, a, false, bb1,
                                                     (short)0, acc1, false, false);
    }

    // D layout: VGPR r -> (M = r + 8*(lane>=16), N = lane&15). Bias + ReLU to LDS.
    const int mBase = (lane >> 4) * 8;
#pragma unroll
    for (int r = 0; r < 8; ++r) {
        hlds[wave][mBase + r][ncol]      = fmaxf(acc0[r] + b1[ncol], 0.0f);
        hlds[wave][mBase + r][ncol + 16] = fmaxf(acc1[r] + b1[ncol + 16], 0.0f);
    }
    __syncthreads();

    // Second (tiny) GEMM [16x32]@[32x8] + row softmax: lanes 0..15 own one row each.
    if (lane < 16) {
        float logit[C];
#pragma unroll
        for (int c = 0; c < C; ++c) logit[c] = b2[c];
        for (int kk = 0; kk < H; ++kk) {
            const float hv = hlds[wave][lane][kk];
#pragma unroll
            for (int c = 0; c < C; ++c)
                logit[c] = fmaf(hv, W2[kk * C + c], logit[c]);
        }
        float mx = logit[0];
#pragma unroll
        for (int c = 1; c < C; ++c) mx = fmaxf(mx, logit[c]);
        float se = 0.0f;
#pragma unroll
        for (int c = 0; c < C; ++c) { logit[c] = __expf(logit[c] - mx); se += logit[c]; }
        const float inv = 1.0f / se;
#pragma unroll
        for (int c = 0; c < C; ++c)
            P0[(size_t)(row0 + lane) * C + c] = logit[c] * inv;
    }
}

// ---------------------------------------------------------------------------
// Kernel 2: one propagation step.
//   pre[i,j]  = ALPHA * (1 - exp2(sum_k log2(1 - min(A[i,k]*P[k,j], 1-eps)))) + BETA*P[i,j]
//   out[i,j]  = softmax_j(pre[i,j]);  Pnext = pre (carried un-softmaxed)
// P is staged transposed in LDS ([C][N] = 64 KB) so the k-streaming inner loop
// is bank-conflict free. One wave per row of A; 8 rows / block; 256 blocks.
// ---------------------------------------------------------------------------
__global__ __launch_bounds__(256) void prop_kernel(
    const float* __restrict__ adj, const float* __restrict__ Pin,
    float* __restrict__ Pnext, float* __restrict__ outS)
{
    __shared__ float Pl[C * N];  // [j][k], 64 KB

    const int t = threadIdx.x;
    for (int f = t; f < N * C; f += 256) {
        const int k = f >> 3, j = f & 7;
        Pl[j * N + k] = Pin[f];
    }
    __syncthreads();

    const int wave = t >> 5;
    const int lane = t & 31;
    const int i = blockIdx.x * 8 + wave;
    const float* Arow = adj + (size_t)i * N;

    float acc[C] = {0.f, 0.f, 0.f, 0.f, 0.f, 0.f, 0.f, 0.f};
    for (int k = lane; k < N; k += 32) {
        const float a = Arow[k];
        __builtin_prefetch(Arow + k + 512, 0, 1);  // global_prefetch_b8, ~16 iters ahead
#pragma unroll
        for (int j = 0; j < C; ++j) {
            const float tt = fminf(a * Pl[j * N + k], ONE_M_EPS);
            acc[j] += __builtin_amdgcn_logf(1.0f - tt);   // v_log_f32 (log2)
        }
    }

    // wave32 reduction
#pragma unroll
    for (int j = 0; j < C; ++j) {
        float v = acc[j];
#pragma unroll
        for (int off = 16; off > 0; off >>= 1) v += __shfl_xor(v, off, 32);
        acc[j] = v;
    }

    if (lane == 0) {
        float pr[C];
        float mx = -1e30f;
#pragma unroll
        for (int j = 0; j < C; ++j) {
            const float orv = 1.0f - __builtin_amdgcn_exp2f(acc[j]);  // v_exp_f32, noisy-OR
            const float pre = ALPHA * orv + BETA * Pl[j * N + i];
            Pnext[(size_t)i * C + j] = pre;
            pr[j] = pre;
            mx = fmaxf(mx, pre);
        }
        float se = 0.0f;
#pragma unroll
        for (int j = 0; j < C; ++j) { pr[j] = __expf(pr[j] - mx); se += pr[j]; }
        const float inv = 1.0f / se;
#pragma unroll
        for (int j = 0; j < C; ++j) outS[(size_t)i * C + j] = pr[j] * inv;
    }
}

// ---------------------------------------------------------------------------
extern "C" void kernel_launch(void* const* d_in, const int* in_sizes, int n_in,
                              void* d_out, int out_size, void* d_ws, size_t ws_size,
                              hipStream_t stream) {
    const float* adj = (const float*)d_in[0];  // [N,N]
    const float* x   = (const float*)d_in[1];  // [N,F]
    const float* W1  = (const float*)d_in[2];  // [F,H]
    const float* b1  = (const float*)d_in[3];  // [H]
    const float* W2  = (const float*)d_in[4];  // [H,C]
    const float* b2  = (const float*)d_in[5];  // [C]
    float* out = (float*)d_out;                // [3,N,C]

    float* Pbuf0 = (float*)d_ws;               // [N,C]
    float* Pbuf1 = Pbuf0 + (size_t)N * C;      // [N,C]

    mlp_softmax_kernel<<<N / (16 * 4), 128, 0, stream>>>(x, W1, b1, W2, b2, Pbuf0);

    float* pin = Pbuf0;
    float* pnx = Pbuf1;
    for (int it = 0; it < 3; ++it) {
        prop_kernel<<<N / 8, 256, 0, stream>>>(adj, pin, pnx, out + (size_t)it * N * C);
        float* tmp = pin; pin = pnx; pnx = tmp;
    }
}